// VectorQuantize_19722489823682
// MI455X (gfx1250) — compile-verified
//
#include <hip/hip_runtime.h>
#include <hip/hip_bf16.h>

typedef __attribute__((ext_vector_type(2))) float v2f;
typedef __attribute__((ext_vector_type(8))) float v8f;

#define VQ_B     16
#define VQ_DIN   512
#define VQ_T     4096
#define VQ_DCB   8
#define VQ_K     1024
#define VQ_TT    256   // timesteps per workgroup

// ---------------- prep: weight_norm for in_project (dim=0 over 512) ----------
__global__ __launch_bounds__(128) void vq_prep_win(const float* __restrict__ in_v,
                                                   const float* __restrict__ in_g,
                                                   float* __restrict__ w_in /* [512*8], [d][c] */) {
  __shared__ float red[128];
  const int c = blockIdx.x;     // 0..7
  const int tid = threadIdx.x;  // 0..127
  float s = 0.f;
  for (int d = tid; d < VQ_DIN; d += 128) { float v = in_v[c * VQ_DIN + d]; s += v * v; }
  red[tid] = s;
  __syncthreads();
  for (int o = 64; o > 0; o >>= 1) {
    if (tid < o) red[tid] += red[tid + o];
    __syncthreads();
  }
  const float inv = in_g[c] / sqrtf(red[0]);
  for (int d = tid; d < VQ_DIN; d += 128) w_in[d * VQ_DCB + c] = in_v[c * VQ_DIN + d] * inv;
}

// -------- prep: weight_norm for out_project + zero commit + -0.5*||code||^2 --
__global__ __launch_bounds__(256) void vq_prep_misc(const float* __restrict__ out_v,
                                                    const float* __restrict__ out_g,
                                                    const float* __restrict__ cb,
                                                    float* __restrict__ w_out /* [512*8] */,
                                                    float* __restrict__ ncnorm /* [1024] */,
                                                    float* __restrict__ commit) {
  const int gid = blockIdx.x * 256 + threadIdx.x;
  if (gid < VQ_DIN) {
    float v[VQ_DCB];
    float s = 0.f;
    for (int c = 0; c < VQ_DCB; ++c) { v[c] = out_v[gid * VQ_DCB + c]; s += v[c] * v[c]; }
    const float inv = out_g[gid] / sqrtf(s);
    for (int c = 0; c < VQ_DCB; ++c) w_out[gid * VQ_DCB + c] = v[c] * inv;
  } else if (gid < VQ_DIN + VQ_B) {
    commit[gid - VQ_DIN] = 0.f;
  } else if (gid < VQ_DIN + VQ_B + VQ_K) {
    const int k = gid - (VQ_DIN + VQ_B);
    float s = 0.f;
    for (int c = 0; c < VQ_DCB; ++c) { float v = cb[k * VQ_DCB + c]; s += v * v; }
    ncnorm[k] = -0.5f * s;
  }
}

// ---------------- main fused kernel ------------------------------------------
__global__ __launch_bounds__(256) void vq_main(const float* __restrict__ z,
                                               const float* __restrict__ in_b,
                                               const float* __restrict__ out_b,
                                               const float* __restrict__ cb,
                                               const float* __restrict__ w_in,
                                               const float* __restrict__ w_out,
                                               const float* __restrict__ ncnorm,
                                               float* __restrict__ out,
                                               float* __restrict__ commit,
                                               int* __restrict__ idx_out) {
  __shared__ float wS[VQ_DIN * VQ_DCB];     // 16 KB: w_in (phase 1), then w_out (phase 3)
  __shared__ float cbS[VQ_K * VQ_DCB];      // 32 KB
  __shared__ float ncS[VQ_K];               // 4 KB: -0.5*||code||^2
  __shared__ float out_bS[VQ_DIN];          // 2 KB
  __shared__ float in_bS[VQ_DCB];
  __shared__ float encS[VQ_TT * VQ_DCB];    // 8 KB, [tLocal][c]
  __shared__ int   idxS[VQ_TT];             // 1 KB

  const int tid = threadIdx.x;
  const int b = blockIdx.x >> 4;            // 16 tiles of 256 per batch element
  const int t0 = (blockIdx.x & 15) * VQ_TT;

  // ---- phase 0: stage parameters ----
  for (int i = tid; i < VQ_DIN * VQ_DCB; i += 256) wS[i] = w_in[i];
  for (int i = tid; i < VQ_K * VQ_DCB; i += 256) cbS[i] = cb[i];
  for (int i = tid; i < VQ_K; i += 256) ncS[i] = ncnorm[i];
  for (int i = tid; i < VQ_DIN; i += 256) out_bS[i] = out_b[i];
  if (tid < VQ_DCB) in_bS[tid] = in_b[tid];
  __syncthreads();

  // ---- phase 1: z_e[c] for t = t0 + tid (stream z, coalesced over tid) ----
  {
    float acc[VQ_DCB];
    for (int c = 0; c < VQ_DCB; ++c) acc[c] = in_bS[c];
    const float* zp = z + (size_t)b * VQ_DIN * VQ_T + (t0 + tid);
    const float4* w4 = (const float4*)wS;
    for (int d = 0; d < VQ_DIN; ++d) {
      if ((d & 63) == 0 && d + 128 < VQ_DIN)
        __builtin_prefetch(&zp[(size_t)(d + 128) * VQ_T], 0, 1);
      const float zv = zp[(size_t)d * VQ_T];
      const float4 lo = w4[d * 2 + 0];
      const float4 hi = w4[d * 2 + 1];
      acc[0] += lo.x * zv; acc[1] += lo.y * zv; acc[2] += lo.z * zv; acc[3] += lo.w * zv;
      acc[4] += hi.x * zv; acc[5] += hi.y * zv; acc[6] += hi.z * zv; acc[7] += hi.w * zv;
    }
    float4* e4 = (float4*)&encS[tid * VQ_DCB];
    e4[0] = make_float4(acc[0], acc[1], acc[2], acc[3]);
    e4[1] = make_float4(acc[4], acc[5], acc[6], acc[7]);
  }
  __syncthreads();

  // reuse wS for w_out (only read in phase 3; phase 2 never touches wS)
  for (int i = tid; i < VQ_DIN * VQ_DCB; i += 256) wS[i] = w_out[i];

  // ---- phase 2: WMMA distance + argmin ----
  // scores = enc(16x8) @ cbT(8x16) via two k=4 f32 WMMAs (C starts at inline 0),
  // then fold -0.5*||code||^2 and track running argmax == argmin distance.
  {
    const int lane = tid & 31;
    const int wave = tid >> 5;
    const int m = lane & 15;     // A row / D column within tile
    const int kh = lane >> 4;    // k-half selector (lanes 0-15: k{0,1}, 16-31: k{2,3})
    const v8f Z = {};            // all-zero C -> inline SRC2 0

    for (int sub = 0; sub < 2; ++sub) {
      const int tb = (wave * 2 + sub) * 16;  // local 16-row tile base
      const v2f* er = (const v2f*)&encS[(tb + m) * VQ_DCB + kh * 2];
      const v2f A0 = er[0];   // enc[t][kh*2 + {0,1}]
      const v2f A1 = er[2];   // enc[t][4 + kh*2 + {0,1}]

      float bestV[8];
      int bestI[8];
      for (int r = 0; r < 8; ++r) { bestV[r] = -3.4e38f; bestI[r] = 0; }

      for (int nb = 0; nb < VQ_K / 16; ++nb) {
        const int col = nb * 16 + m;
        const v2f* cr = (const v2f*)&cbS[col * VQ_DCB + kh * 2];
        const v2f B0 = cr[0];
        const v2f B1 = cr[2];
        const float nc = ncS[col];
        v8f D = __builtin_amdgcn_wmma_f32_16x16x4_f32(false, A0, false, B0, (short)0, Z,
                                                      false, false);
        D = __builtin_amdgcn_wmma_f32_16x16x4_f32(false, A1, false, B1, (short)0, D,
                                                  false, false);
        for (int r = 0; r < 8; ++r) {
          const float v = D[r] + nc;
          const bool p = v > bestV[r];
          bestV[r] = p ? v : bestV[r];
          bestI[r] = p ? col : bestI[r];
        }
      }
      // reduce over the 16 lanes of each half (D layout: lane%16 = column,
      // vgpr slot r = row r + 8*(lane>=16)); masks <16 stay inside the half.
      for (int off = 1; off < 16; off <<= 1) {
        for (int r = 0; r < 8; ++r) {
          const float ov = __shfl_xor(bestV[r], off, 32);
          const int oi = __shfl_xor(bestI[r], off, 32);
          const bool p = (ov > bestV[r]) || (ov == bestV[r] && oi < bestI[r]);
          bestV[r] = p ? ov : bestV[r];
          bestI[r] = p ? oi : bestI[r];
        }
      }
      if (m == 0) {  // lane 0 -> rows 0..7, lane 16 -> rows 8..15
        for (int r = 0; r < 8; ++r) idxS[tb + r + 8 * kh] = bestI[r];
      }
    }
  }
  __syncthreads();

  // ---- phase 3: gather z_q, commit loss, out-projection ----
  {
    const int t = t0 + tid;
    const int k = idxS[tid];
    idx_out[b * VQ_T + t] = k;

    const float4* cb4 = (const float4*)cbS;
    const float4 q0 = cb4[k * 2 + 0];
    const float4 q1 = cb4[k * 2 + 1];
    const float4* e4 = (const float4*)&encS[tid * VQ_DCB];
    const float4 e0 = e4[0], e1 = e4[1];

    float cm = 0.f, dx;
    dx = e0.x - q0.x; cm += dx * dx;  dx = e0.y - q0.y; cm += dx * dx;
    dx = e0.z - q0.z; cm += dx * dx;  dx = e0.w - q0.w; cm += dx * dx;
    dx = e1.x - q1.x; cm += dx * dx;  dx = e1.y - q1.y; cm += dx * dx;
    dx = e1.z - q1.z; cm += dx * dx;  dx = e1.w - q1.w; cm += dx * dx;

    float* op = out + (size_t)b * VQ_DIN * VQ_T + t;
    const float4* w4 = (const float4*)wS;
    for (int d = 0; d < VQ_DIN; ++d) {
      const float4 wl = w4[d * 2 + 0];
      const float4 wh = w4[d * 2 + 1];
      const float a = wl.x * q0.x + wl.y * q0.y + wl.z * q0.z + wl.w * q0.w +
                      wh.x * q1.x + wh.y * q1.y + wh.z * q1.z + wh.w * q1.w + out_bS[d];
      op[(size_t)d * VQ_T] = a;
    }

    for (int off = 16; off > 0; off >>= 1) cm += __shfl_xor(cm, off, 32);
    if ((tid & 31) == 0)
      atomicAdd(&commit[b], cm * (1.0f / (float)(VQ_DCB * VQ_T)));
  }
}

extern "C" void kernel_launch(void* const* d_in, const int* in_sizes, int n_in,
                              void* d_out, int out_size, void* d_ws, size_t ws_size,
                              hipStream_t stream) {
  (void)in_sizes; (void)n_in; (void)out_size; (void)ws_size;
  const float* z     = (const float*)d_in[0];
  const float* in_v  = (const float*)d_in[1];
  const float* in_g  = (const float*)d_in[2];
  const float* in_b  = (const float*)d_in[3];
  const float* out_v = (const float*)d_in[4];
  const float* out_g = (const float*)d_in[5];
  const float* out_b = (const float*)d_in[6];
  const float* cb    = (const float*)d_in[7];

  float* ws     = (float*)d_ws;
  float* w_in   = ws;                               // [512*8]
  float* w_out  = ws + VQ_DIN * VQ_DCB;             // [512*8]
  float* ncnorm = ws + 2 * VQ_DIN * VQ_DCB;         // [1024]

  float* out    = (float*)d_out;                                   // 16*512*4096
  float* commit = out + (size_t)VQ_B * VQ_DIN * VQ_T;              // 16
  int*   idx    = (int*)(commit + VQ_B);                           // 16*4096

  vq_prep_win<<<VQ_DCB, 128, 0, stream>>>(in_v, in_g, w_in);
  vq_prep_misc<<<7, 256, 0, stream>>>(out_v, out_g, cb, w_out, ncnorm, commit);
  vq_main<<<VQ_B * (VQ_T / VQ_TT), 256, 0, stream>>>(z, in_b, out_b, cb, w_in, w_out,
                                                     ncnorm, out, commit, idx);
}